// SACNN_69569880261141
// MI455X (gfx1250) — compile-verified
//
#include <hip/hip_runtime.h>

typedef float v2f __attribute__((ext_vector_type(2)));
typedef float v8f __attribute__((ext_vector_type(8)));

#define HWp   2304      // 48*48
#define WDIM  48
#define CDIM  256
#define NDIM  2
#define BNS   0.9999950000374997f   // 1/sqrt(1+1e-5)

__device__ __forceinline__ int reflect48(int i) {
    i = (i < 0) ? -i : i;
    return (i >= WDIM) ? (2 * WDIM - 2 - i) : i;
}

// ---------------------------------------------------------------------------
// Kernel 1: per-(n,c) spatial mean; x0 = (x*mean)*g0*BNS + b0  -> cat[ch 0..255]
// grid: (C, N), block 256
// ---------------------------------------------------------------------------
__global__ __launch_bounds__(256)
void mean_x0_kernel(const float* __restrict__ x, const float* __restrict__ g0,
                    const float* __restrict__ b0, float* __restrict__ cat) {
    const int n = blockIdx.y, c = blockIdx.x, t = threadIdx.x;
    const float* xb = x + ((long)n * CDIM + c) * HWp;
    __shared__ float red[256];
    float s = 0.f;
    for (int p = t; p < HWp; p += 256) s += xb[p];
    red[t] = s;
    __syncthreads();
    for (int o = 128; o > 0; o >>= 1) {
        if (t < o) red[t] += red[t + o];
        __syncthreads();
    }
    const float mean = red[0] * (1.0f / HWp);
    const float sc = mean * g0[c] * BNS;
    const float bo = b0[c];
    float* cb = cat + ((long)n * 1024 + c) * HWp;
    for (int p = t; p < HWp; p += 256) cb[p] = xb[p] * sc + bo;
}

// ---------------------------------------------------------------------------
// Kernel 2: per-position channel max & mean
// ---------------------------------------------------------------------------
__global__ __launch_bounds__(256)
void maxmean_kernel(const float* __restrict__ x, float* __restrict__ xmax,
                    float* __restrict__ xmean) {
    const int idx = blockIdx.x * 256 + threadIdx.x;
    if (idx >= NDIM * HWp) return;
    const int n = idx / HWp, p = idx - n * HWp;
    const float* xb = x + (long)n * CDIM * HWp + p;
    float mx = -1e30f, s = 0.f;
    for (int c = 0; c < CDIM; ++c) {
        const float v = xb[(long)c * HWp];
        mx = fmaxf(mx, v);
        s += v;
    }
    xmax[idx] = mx;
    xmean[idx] = s * (1.0f / CDIM);
}

// ---------------------------------------------------------------------------
// Kernel 3: generic fp32 WMMA GEMM  Out[m,p] = sum_k W[m,k]*X[k,p] + bias[m]
// grid: (M/128, HWp/64, N). block 256 (8 waves); BM=128, BN=64, BK=64.
// Each wave: 16x64 (4 accumulators) -> 4x A-fragment reuse.
// A LDS stride 68 (float4-aligned, conflict-free b64 fragment reads).
// B LDS pair-interleaved: B[kk][col] at q*160 + col*2 + (kk&1), q=kk/2;
// stride 160 => half0/half1 lane groups hit disjoint bank halves.
// ---------------------------------------------------------------------------
__global__ __launch_bounds__(256)
void gemm_wmma_kernel(const float* __restrict__ Wm, const float* __restrict__ X,
                      const float* __restrict__ bias, float* __restrict__ Out,
                      int M, int K, long Xn_stride, long On_stride) {
    constexpr int AS = 68;
    constexpr int BSs = 160;
    __shared__ float As[128 * AS];
    __shared__ float Bs[32 * BSs];
    const int t = threadIdx.x;
    const int wv = t >> 5;
    const int lane = t & 31;
    const int l15 = lane & 15;
    const int half = lane >> 4;
    const int m0 = blockIdx.x * 128;
    const int p0 = blockIdx.y * 64;
    const float* Xb = X + (long)blockIdx.z * Xn_stride;
    float* Ob = Out + (long)blockIdx.z * On_stride;

    v8f acc0 = {}, acc1 = {}, acc2 = {}, acc3 = {};
    for (int kb = 0; kb < K; kb += 64) {
        #pragma unroll
        for (int i = 0; i < 8; ++i) {               // A: 128x64, float4
            const int e = t + 256 * i;              // float4 index
            const int r = e >> 4, q4 = e & 15;
            const float4 v = *(const float4*)(Wm + (long)(m0 + r) * K + kb + q4 * 4);
            *(float4*)(&As[r * AS + q4 * 4]) = v;
        }
        #pragma unroll
        for (int i = 0; i < 4; ++i) {               // B: 64x64, float4 -> paired
            const int e = t + 256 * i;
            const int r = e >> 4, c4 = e & 15;
            const float4 v = *(const float4*)(Xb + (long)(kb + r) * HWp + p0 + c4 * 4);
            float* dst = &Bs[(r >> 1) * BSs + (r & 1)];
            dst[(c4 * 4 + 0) * 2] = v.x;
            dst[(c4 * 4 + 1) * 2] = v.y;
            dst[(c4 * 4 + 2) * 2] = v.z;
            dst[(c4 * 4 + 3) * 2] = v.w;
        }
        __syncthreads();
        const float* Aw = As + (wv * 16 + l15) * AS;
        #pragma unroll
        for (int s = 0; s < 16; ++s) {
            const int kk0 = s * 4 + 2 * half;
            const v2f a = *(const v2f*)(Aw + kk0);
            const float* Bq = Bs + (2 * s + half) * BSs + l15 * 2;
            const v2f b0 = *(const v2f*)(Bq + 0);
            const v2f b1 = *(const v2f*)(Bq + 32);
            const v2f b2 = *(const v2f*)(Bq + 64);
            const v2f b3 = *(const v2f*)(Bq + 96);
            acc0 = __builtin_amdgcn_wmma_f32_16x16x4_f32(false, a, false, b0, (short)0, acc0, false, false);
            acc1 = __builtin_amdgcn_wmma_f32_16x16x4_f32(false, a, false, b1, (short)0, acc1, false, false);
            acc2 = __builtin_amdgcn_wmma_f32_16x16x4_f32(false, a, false, b2, (short)0, acc2, false, false);
            acc3 = __builtin_amdgcn_wmma_f32_16x16x4_f32(false, a, false, b3, (short)0, acc3, false, false);
        }
        __syncthreads();
    }
    #pragma unroll
    for (int v = 0; v < 8; ++v) {
        const int m = m0 + wv * 16 + v + 8 * half;
        const float bi = bias[m];
        float* orow = Ob + (long)m * HWp + p0;
        orow[l15 +  0] = acc0[v] + bi;
        orow[l15 + 16] = acc1[v] + bi;
        orow[l15 + 32] = acc2[v] + bi;
        orow[l15 + 48] = acc3[v] + bi;
    }
}

// ---------------------------------------------------------------------------
// Kernel 4: sigma conv as im2col WMMA GEMM + BN + softmax over k^2 channels
// A = Wsig (k^2 x C*k^2) zero-padded to MT*16 rows; B gathered with reflect pad
// grid: (HWp/BC, N), block 256
// ---------------------------------------------------------------------------
template <int KSZ>
__global__ __launch_bounds__(256)
void sigma_kernel(const float* __restrict__ Wsig, const float* __restrict__ g,
                  const float* __restrict__ bb, const float* __restrict__ x,
                  float* __restrict__ sig) {
    constexpr int K2 = KSZ * KSZ;
    constexpr int MT = (K2 + 15) / 16;     // 1,2,4
    constexpr int NW = 8 / MT;             // col subtiles per block
    constexpr int BC = NW * 16;            // block columns: 128,64,32
    constexpr int KTOT = CDIM * K2;        // 2304,6400,12544 (all %64==0)
    constexpr int PD = KSZ / 2;

    __shared__ float As[MT * 16 * 66];
    __shared__ float Bs[64 * BC];          // reused to stage D for softmax

    const int t = threadIdx.x;
    const int wv = t >> 5;
    const int lane = t & 31;
    const int l15 = lane & 15;
    const int half = lane >> 4;
    const int n = blockIdx.y;
    const int pbase = blockIdx.x * BC;
    const float* xb = x + (long)n * CDIM * HWp;
    const int mi = wv / NW, ci = wv % NW;

    v8f acc = {};
    for (int kb = 0; kb < KTOT; kb += 64) {
        for (int e = t; e < MT * 16 * 64; e += 256) {      // A with zero-pad rows
            const int r = e >> 6, kk = e & 63;
            As[r * 66 + kk] = (r < K2) ? Wsig[(long)r * KTOT + kb + kk] : 0.0f;
        }
        for (int e = t; e < 64 * BC; e += 256) {           // B im2col gather
            const int rr = e / BC, col = e - rr * BC;
            const int kk = kb + rr;
            const int c = kk / K2;
            const int rem = kk - c * K2;
            const int dy = rem / KSZ;
            const int dx = rem - dy * KSZ;
            const int p = pbase + col;
            const int h = p / WDIM, w = p - h * WDIM;
            const int hs = reflect48(h + dy - PD);
            const int ws = reflect48(w + dx - PD);
            Bs[e] = xb[(long)c * HWp + hs * WDIM + ws];
        }
        __syncthreads();
        const float* Aw = As + (mi * 16 + l15) * 66;
        #pragma unroll
        for (int s = 0; s < 16; ++s) {
            const int kk0 = s * 4 + 2 * half;
            v2f a; a.x = Aw[kk0];                       a.y = Aw[kk0 + 1];
            v2f b; b.x = Bs[kk0 * BC + ci * 16 + l15];  b.y = Bs[(kk0 + 1) * BC + ci * 16 + l15];
            acc = __builtin_amdgcn_wmma_f32_16x16x4_f32(
                false, a, false, b, (short)0, acc, false, false);
        }
        __syncthreads();
    }
    #pragma unroll
    for (int v = 0; v < 8; ++v) {                       // stage D into LDS
        const int r = mi * 16 + v + 8 * half;
        Bs[r * BC + ci * 16 + l15] = acc[v];
    }
    __syncthreads();
    if (t < BC) {                                       // BN + softmax per column
        float mx = -1e30f;
        for (int j = 0; j < K2; ++j)
            mx = fmaxf(mx, Bs[j * BC + t] * (g[j] * BNS) + bb[j]);
        float sum = 0.f;
        for (int j = 0; j < K2; ++j)
            sum += expf(Bs[j * BC + t] * (g[j] * BNS) + bb[j] - mx);
        const float inv = 1.0f / sum;
        const int p = pbase + t;
        for (int j = 0; j < K2; ++j)
            sig[((long)n * 64 + j) * HWp + p] =
                expf(Bs[j * BC + t] * (g[j] * BNS) + bb[j] - mx) * inv;
    }
}

// ---------------------------------------------------------------------------
// Kernel 5: branch combine: out[c,p] = gate(p) * sum_j ypad[c,src_j]*sig[j,p]
// grid: (HWp/32, N), block 256 (8 channel-groups x 32 positions)
// ---------------------------------------------------------------------------
template <int KSZ>
__global__ __launch_bounds__(256)
void combine_kernel(const float* __restrict__ y, const float* __restrict__ sig,
                    const float* __restrict__ xmax, const float* __restrict__ xmean,
                    float* __restrict__ cat, int coff) {
    constexpr int K2 = KSZ * KSZ;
    constexpr int PD = KSZ / 2;
    constexpr int TP = 32;
    __shared__ float sS[K2 * TP];
    __shared__ int   sI[K2 * TP];
    __shared__ float sG[TP];
    const int n = blockIdx.y;
    const int pbase = blockIdx.x * TP;
    const int t = threadIdx.x;

    for (int e = t; e < K2 * TP; e += 256) {
        const int j = e / TP, pl = e - j * TP;
        const int p = pbase + pl;
        const int h = p / WDIM, w = p - h * WDIM;
        const int dy = j / KSZ, dx = j - dy * KSZ;
        sI[e] = reflect48(h + dy - PD) * WDIM + reflect48(w + dx - PD);
        sS[e] = sig[((long)n * 64 + j) * HWp + p];
    }
    __syncthreads();
    if (t < TP) {
        const float* xm = xmax + (long)n * HWp;
        const float* xe = xmean + (long)n * HWp;
        float a = 0.f;
        for (int j = 0; j < K2; ++j) {
            const int s = sI[j * TP + t];
            a += (xm[s] + xe[s]) * sS[j * TP + t];
        }
        sG[t] = 1.0f / (1.0f + expf(-a));
    }
    __syncthreads();
    const int pl = t & 31;
    const int cgrp = t >> 5;
    const float gate = sG[pl];
    const float* yb = y + (long)n * CDIM * HWp;
    float* cb = cat + ((long)n * 1024 + coff) * HWp;
    for (int c = cgrp; c < CDIM; c += 8) {
        const float* yc = yb + (long)c * HWp;
        float a = 0.f;
        for (int j = 0; j < K2; ++j)
            a += yc[sI[j * TP + pl]] * sS[j * TP + pl];
        cb[(long)c * HWp + pbase + pl] = a * gate;
    }
}

// ---------------------------------------------------------------------------
extern "C" void kernel_launch(void* const* d_in, const int* in_sizes, int n_in,
                              void* d_out, int out_size, void* d_ws, size_t ws_size,
                              hipStream_t stream) {
    (void)in_sizes; (void)n_in; (void)out_size; (void)ws_size;
    const float* x     = (const float*)d_in[0];
    const float* Wsig1 = (const float*)d_in[1];
    const float* g1    = (const float*)d_in[2];
    const float* b1    = (const float*)d_in[3];
    const float* Wc1_1 = (const float*)d_in[4];
    const float* bc1_1 = (const float*)d_in[5];
    const float* Wsig2 = (const float*)d_in[6];
    const float* g2    = (const float*)d_in[7];
    const float* b2    = (const float*)d_in[8];
    const float* Wc1_2 = (const float*)d_in[9];
    const float* bc1_2 = (const float*)d_in[10];
    const float* Wsig3 = (const float*)d_in[11];
    const float* g3    = (const float*)d_in[12];
    const float* b3    = (const float*)d_in[13];
    const float* Wc1_3 = (const float*)d_in[14];
    const float* bc1_3 = (const float*)d_in[15];
    const float* g0    = (const float*)d_in[16];
    const float* b0    = (const float*)d_in[17];
    const float* Wout  = (const float*)d_in[18];
    const float* bout  = (const float*)d_in[19];
    float* out = (float*)d_out;

    float* cat   = (float*)d_ws;                               // N*1024*HWp
    float* ybuf  = cat  + (size_t)NDIM * 1024 * HWp;           // N*256*HWp
    float* sig   = ybuf + (size_t)NDIM * CDIM * HWp;           // N*64*HWp
    float* xmax  = sig  + (size_t)NDIM * 64 * HWp;             // N*HWp
    float* xmean = xmax + (size_t)NDIM * HWp;                  // N*HWp

    mean_x0_kernel<<<dim3(CDIM, NDIM), 256, 0, stream>>>(x, g0, b0, cat);
    maxmean_kernel<<<dim3((NDIM * HWp + 255) / 256), 256, 0, stream>>>(x, xmax, xmean);

    // branch k=3 -> cat channels [256,512)
    sigma_kernel<3><<<dim3(HWp / 128, NDIM), 256, 0, stream>>>(Wsig1, g1, b1, x, sig);
    gemm_wmma_kernel<<<dim3(2, HWp / 64, NDIM), 256, 0, stream>>>(
        Wc1_1, x, bc1_1, ybuf, 256, 256, (long)CDIM * HWp, (long)CDIM * HWp);
    combine_kernel<3><<<dim3(HWp / 32, NDIM), 256, 0, stream>>>(ybuf, sig, xmax, xmean, cat, 256);

    // branch k=5 -> cat channels [512,768)
    sigma_kernel<5><<<dim3(HWp / 64, NDIM), 256, 0, stream>>>(Wsig2, g2, b2, x, sig);
    gemm_wmma_kernel<<<dim3(2, HWp / 64, NDIM), 256, 0, stream>>>(
        Wc1_2, x, bc1_2, ybuf, 256, 256, (long)CDIM * HWp, (long)CDIM * HWp);
    combine_kernel<5><<<dim3(HWp / 32, NDIM), 256, 0, stream>>>(ybuf, sig, xmax, xmean, cat, 512);

    // branch k=7 -> cat channels [768,1024)
    sigma_kernel<7><<<dim3(HWp / 32, NDIM), 256, 0, stream>>>(Wsig3, g3, b3, x, sig);
    gemm_wmma_kernel<<<dim3(2, HWp / 64, NDIM), 256, 0, stream>>>(
        Wc1_3, x, bc1_3, ybuf, 256, 256, (long)CDIM * HWp, (long)CDIM * HWp);
    combine_kernel<7><<<dim3(HWp / 32, NDIM), 256, 0, stream>>>(ybuf, sig, xmax, xmean, cat, 768);

    // final: out = Wout(2048x1024) * cat + bout
    gemm_wmma_kernel<<<dim3(16, HWp / 64, NDIM), 256, 0, stream>>>(
        Wout, cat, bout, out, 2048, 1024, (long)1024 * HWp, (long)2048 * HWp);
}